// GNN_7962869367464
// MI455X (gfx1250) — compile-verified
//
#include <hip/hip_runtime.h>

// ---------------------------------------------------------------------------
// 2-layer GCN for MI455X (gfx1250), wave32.
//   layer(x, W, b): h = x @ W           (WMMA f32 16x16x4)
//                   deg = 1 + sum_in ew ; dinv = rsqrt(deg)
//                   out[i] = b + dinv[i]^2 * h[i] + sum_{(j->i)} dinv[j]*ew*dinv[i]*h[j]
//   out = layer2(relu(layer1(x)))
// ---------------------------------------------------------------------------

#define N_NODES 100000
#define N_EDGES 1000000
#define HIDDEN  64
#define ROW_TILES (N_NODES / 16)   // 6250 exactly

typedef __attribute__((ext_vector_type(2))) float v2f;
typedef __attribute__((ext_vector_type(8))) float v8f;

// ---------------- degree / norm helpers ----------------

__global__ void k_fill1(float* deg, int n) {
    int i = blockIdx.x * blockDim.x + threadIdx.x;
    if (i < n) deg[i] = 1.0f;               // self-loop weight
}

__global__ void k_deg_accum(float* deg, const int* dst, const float* ew, int e) {
    int i = blockIdx.x * blockDim.x + threadIdx.x;
    if (i < e) atomicAdd(&deg[dst[i]], ew[i]);
}

__global__ void k_dinv(float* deg, int n) {
    int i = blockIdx.x * blockDim.x + threadIdx.x;
    if (i < n) {
        float d = deg[i];
        deg[i] = (d > 0.0f) ? __frsqrt_rn(d) : 0.0f;
    }
}

// ---------------- dense projection: Y = X @ W  (N x 64) ----------------
// One wave per 16-row strip; 4 col-tiles of 16; K = 64 in 16 steps of 4.
// Uses V_WMMA_F32_16X16X4_F32 (full fp32 path of CDNA5).

__global__ void k_gemm_wmma(const float* __restrict__ X,
                            const float* __restrict__ W,
                            float* __restrict__ Y,
                            int nTiles) {
    const int wavesPerBlock = blockDim.x >> 5;
    const int m = blockIdx.x * wavesPerBlock + ((int)threadIdx.x >> 5);
    if (m >= nTiles) return;                 // uniform per-wave exit: EXEC stays all-1
    const int lane  = threadIdx.x & 31;
    const int l16   = lane & 15;
    const int khalf = (lane >> 4) << 1;      // 0 or 2

    v8f acc0 = {}, acc1 = {}, acc2 = {}, acc3 = {};
    const int row = m * 16 + l16;

    #pragma unroll
    for (int k = 0; k < HIDDEN; k += 4) {
        // A fragment: 16x4, row-major X
        v2f a = *(const v2f*)&X[row * HIDDEN + k + khalf];
        // B fragments: 4x16 slices of row-major W (64x64 = 16KB, cache resident)
        const float* w0 = &W[(k + khalf) * HIDDEN];
        const float* w1 = &W[(k + khalf + 1) * HIDDEN];
        v2f b0 = { w0[ 0 + l16], w1[ 0 + l16] };
        v2f b1 = { w0[16 + l16], w1[16 + l16] };
        v2f b2 = { w0[32 + l16], w1[32 + l16] };
        v2f b3 = { w0[48 + l16], w1[48 + l16] };
        acc0 = __builtin_amdgcn_wmma_f32_16x16x4_f32(false, a, false, b0, (short)0, acc0, false, false);
        acc1 = __builtin_amdgcn_wmma_f32_16x16x4_f32(false, a, false, b1, (short)0, acc1, false, false);
        acc2 = __builtin_amdgcn_wmma_f32_16x16x4_f32(false, a, false, b2, (short)0, acc2, false, false);
        acc3 = __builtin_amdgcn_wmma_f32_16x16x4_f32(false, a, false, b3, (short)0, acc3, false, false);
    }

    // C/D layout: VGPR i -> M = (lane>=16 ? 8 : 0) + i, N = lane&15
    const int rbase = m * 16 + ((lane >> 4) << 3);
    #pragma unroll
    for (int i = 0; i < 8; ++i) {
        float* yr = &Y[(rbase + i) * HIDDEN];
        yr[ 0 + l16] = acc0[i];
        yr[16 + l16] = acc1[i];
        yr[32 + l16] = acc2[i];
        yr[48 + l16] = acc3[i];
    }
}

// ---------------- aggregation ----------------

// acc[i][c] = bias[c] + dinv[i]^2 * h[i][c]   (self-loop term + bias)
__global__ void k_agg_init(float* __restrict__ acc,
                           const float* __restrict__ h,
                           const float* __restrict__ dinv,
                           const float* __restrict__ bias,
                           int n) {
    int t = blockIdx.x * blockDim.x + threadIdx.x;
    if (t < n * HIDDEN) {
        int i = t >> 6;
        int c = t & 63;
        float di = dinv[i];
        acc[t] = bias[c] + di * di * h[t];
    }
}

// one wave per edge; each lane owns 2 columns (float2)
__global__ void k_edge_scatter(float* __restrict__ acc,
                               const float* __restrict__ h,
                               const float* __restrict__ dinv,
                               const int* __restrict__ src,
                               const int* __restrict__ dst,
                               const float* __restrict__ ew,
                               int e) {
    int gid  = blockIdx.x * blockDim.x + threadIdx.x;
    int edge = gid >> 5;
    int lane = gid & 31;
    if (edge >= e) return;
    int s = src[edge];
    int d = dst[edge];
    float norm = dinv[s] * ew[edge] * dinv[d];
    const float2 v = *(const float2*)&h[s * HIDDEN + lane * 2];
    float* out = &acc[d * HIDDEN + lane * 2];
    atomicAdd(&out[0], v.x * norm);
    atomicAdd(&out[1], v.y * norm);
}

__global__ void k_relu(float* x, int n) {
    int i = blockIdx.x * blockDim.x + threadIdx.x;
    if (i < n) x[i] = fmaxf(x[i], 0.0f);
}

// ---------------------------------------------------------------------------

extern "C" void kernel_launch(void* const* d_in, const int* in_sizes, int n_in,
                              void* d_out, int out_size, void* d_ws, size_t ws_size,
                              hipStream_t stream) {
    const float* x   = (const float*)d_in[0];
    const int*   ei  = (const int*)  d_in[1];   // (2, E) flat: src then dst
    const float* ew  = (const float*)d_in[2];
    const float* W1  = (const float*)d_in[3];
    const float* b1  = (const float*)d_in[4];
    const float* W2  = (const float*)d_in[5];
    const float* b2  = (const float*)d_in[6];
    float* out = (float*)d_out;

    const int* src = ei;
    const int* dst = ei + N_EDGES;

    // workspace carve-up (float-aligned)
    float* deg  = (float*)d_ws;                       // N      (becomes dinv)
    float* proj = deg + ((N_NODES + 63) & ~63);       // N * 64 (projection)
    float* act  = proj + N_NODES * HIDDEN;            // N * 64 (layer-1 activations)

    const int B = 256;
    const int gNodes  = (N_NODES + B - 1) / B;
    const int gEdges  = (N_EDGES + B - 1) / B;
    const int gFeat   = (N_NODES * HIDDEN + B - 1) / B;
    const int gEdgeWv = (N_EDGES * 32 + B - 1) / B;       // one wave per edge
    const int gGemm   = (ROW_TILES + (B / 32) - 1) / (B / 32);

    // --- degrees / dinv (shared by both layers) ---
    k_fill1<<<gNodes, B, 0, stream>>>(deg, N_NODES);
    k_deg_accum<<<gEdges, B, 0, stream>>>(deg, dst, ew, N_EDGES);
    k_dinv<<<gNodes, B, 0, stream>>>(deg, N_NODES);

    // --- layer 1 ---
    k_gemm_wmma<<<gGemm, B, 0, stream>>>(x, W1, proj, ROW_TILES);
    k_agg_init<<<gFeat, B, 0, stream>>>(act, proj, deg, b1, N_NODES);
    k_edge_scatter<<<gEdgeWv, B, 0, stream>>>(act, proj, deg, src, dst, ew, N_EDGES);
    k_relu<<<gFeat, B, 0, stream>>>(act, N_NODES * HIDDEN);

    // --- layer 2 (aggregate straight into d_out) ---
    k_gemm_wmma<<<gGemm, B, 0, stream>>>(act, W2, proj, ROW_TILES);
    k_agg_init<<<gFeat, B, 0, stream>>>(out, proj, deg, b2, N_NODES);
    k_edge_scatter<<<gEdgeWv, B, 0, stream>>>(out, proj, deg, src, dst, ew, N_EDGES);
}